// EsmContrastiveProjectionHead_47038481826368
// MI455X (gfx1250) — compile-verified
//
#include <hip/hip_runtime.h>

#define Bsz 512
#define Ssz 512
#define Dsz 960
#define D1  480
#define D2  240
#define SSPLIT 4
#define NEG_INF_V (-65504.0f)
#define INV_T 10.0f
#define EPSV 1e-8f

typedef float v2f __attribute__((ext_vector_type(2)));
typedef float v8f __attribute__((ext_vector_type(8)));

// ---------------------------------------------------------------------------
// D = A(16x4) * B(4x16) + C   (fp32 WMMA, wave32)
// ---------------------------------------------------------------------------
__device__ __forceinline__ v8f wmma_k4(v2f a, v2f b, v8f c) {
  return __builtin_amdgcn_wmma_f32_16x16x4_f32(
      /*neg_a=*/false, a, /*neg_b=*/false, b,
      /*c_mod=*/(short)0, c, /*reuse_a=*/false, /*reuse_b=*/false);
}

// ---------------------------------------------------------------------------
// Phase 1a: partial mean-pool. grid (512, 4), block 240. float4 streaming.
// partial[chunk][b][d] = sum over 128 s of x[b][s][d]
// ---------------------------------------------------------------------------
__global__ __launch_bounds__(240) void pool_partial_kernel(
    const float* __restrict__ x, float* __restrict__ partial) {
  const int b = blockIdx.x;
  const int chunk = blockIdx.y;
  const int t = threadIdx.x;                       // 0..239 (float4 column)
  const float4* __restrict__ xb =
      reinterpret_cast<const float4*>(x) + (size_t)b * (size_t)Ssz * (Dsz / 4);
  const int sBeg = chunk * (Ssz / SSPLIT);
  float4 acc = make_float4(0.f, 0.f, 0.f, 0.f);
#pragma unroll 4
  for (int s = 0; s < Ssz / SSPLIT; ++s) {
    float4 v = xb[(size_t)(sBeg + s) * (Dsz / 4) + t];
    acc.x += v.x; acc.y += v.y; acc.z += v.z; acc.w += v.w;
  }
  reinterpret_cast<float4*>(partial)[((size_t)chunk * Bsz + b) * (Dsz / 4) + t] = acc;
}

// ---------------------------------------------------------------------------
// Phase 1b: reduce 4 partials -> pooled (scaled by 1/S). Deterministic order.
// ---------------------------------------------------------------------------
__global__ __launch_bounds__(256) void pool_reduce_kernel(
    const float* __restrict__ partial, float* __restrict__ pooled) {
  const int idx = blockIdx.x * 256 + threadIdx.x;  // float4 index, 0..122879
  const int n4 = Bsz * Dsz / 4;
  const float4* __restrict__ p4 = reinterpret_cast<const float4*>(partial);
  float4 a = make_float4(0.f, 0.f, 0.f, 0.f);
#pragma unroll
  for (int c = 0; c < SSPLIT; ++c) {
    float4 v = p4[(size_t)c * n4 + idx];
    a.x += v.x; a.y += v.y; a.z += v.z; a.w += v.w;
  }
  const float sc = 1.0f / (float)Ssz;
  a.x *= sc; a.y *= sc; a.z *= sc; a.w *= sc;
  reinterpret_cast<float4*>(pooled)[idx] = a;
}

// ---------------------------------------------------------------------------
// Phase 2: layer-1 GEMM + bias + ReLU (per-half weights), WMMA f32 16x16x4.
// pooled[512,960] x W1[960,480] -> h[512,480]. One wave per 16x16 tile.
// tiles: 32 (M) x 30 (N) = 960; 120 blocks x 8 waves.
// ---------------------------------------------------------------------------
__global__ __launch_bounds__(256) void mlp1_kernel(
    const float* __restrict__ pooled,
    const float* __restrict__ W1c, const float* __restrict__ b1c,
    const float* __restrict__ W1a, const float* __restrict__ b1a,
    float* __restrict__ h) {
  const int w = threadIdx.x >> 5, lane = threadIdx.x & 31;
  const int tile = blockIdx.x * 8 + w;             // 0..959
  const int tm = tile / 30, tn = tile % 30;
  const int m0 = tm * 16, n0 = tn * 16;
  const bool hiHalf = lane >= 16;
  const int row = lane & 15;
  const int koff = hiHalf ? 2 : 0;
  const float* __restrict__ W    = (m0 < Bsz / 2) ? W1c : W1a;
  const float* __restrict__ bias = (m0 < Bsz / 2) ? b1c : b1a;
  const float* __restrict__ Arow = pooled + (size_t)(m0 + row) * Dsz + koff;
  const float* __restrict__ Bcol = W + n0 + row;
  v8f c = {};
#pragma unroll 2
  for (int k = 0; k < Dsz; k += 4) {
    v2f a = *(const v2f*)(Arow + k);
    v2f b;
    b[0] = Bcol[(size_t)(k + koff) * D1];
    b[1] = Bcol[(size_t)(k + koff + 1) * D1];
    c = wmma_k4(a, b, c);
  }
  const int n = n0 + row;
  const float bv = bias[n];
#pragma unroll
  for (int r = 0; r < 8; ++r) {
    const int m = m0 + r + (hiHalf ? 8 : 0);
    float v = c[r] + bv;
    h[(size_t)m * D1 + n] = v > 0.f ? v : 0.f;
  }
}

// ---------------------------------------------------------------------------
// Phase 3: layer-2 GEMM + bias. h[512,480] x W2[480,240] -> z[512,240].
// tiles: 32 x 15 = 480; 60 blocks x 8 waves.
// ---------------------------------------------------------------------------
__global__ __launch_bounds__(256) void mlp2_kernel(
    const float* __restrict__ h,
    const float* __restrict__ W2c, const float* __restrict__ b2c,
    const float* __restrict__ W2a, const float* __restrict__ b2a,
    float* __restrict__ z) {
  const int w = threadIdx.x >> 5, lane = threadIdx.x & 31;
  const int tile = blockIdx.x * 8 + w;             // 0..479
  const int tm = tile / 15, tn = tile % 15;
  const int m0 = tm * 16, n0 = tn * 16;
  const bool hiHalf = lane >= 16;
  const int row = lane & 15;
  const int koff = hiHalf ? 2 : 0;
  const float* __restrict__ W    = (m0 < Bsz / 2) ? W2c : W2a;
  const float* __restrict__ bias = (m0 < Bsz / 2) ? b2c : b2a;
  const float* __restrict__ Arow = h + (size_t)(m0 + row) * D1 + koff;
  const float* __restrict__ Bcol = W + n0 + row;
  v8f c = {};
#pragma unroll 2
  for (int k = 0; k < D1; k += 4) {
    v2f a = *(const v2f*)(Arow + k);
    v2f b;
    b[0] = Bcol[(size_t)(k + koff) * D2];
    b[1] = Bcol[(size_t)(k + koff + 1) * D2];
    c = wmma_k4(a, b, c);
  }
  const int n = n0 + row;
  const float bv = bias[n];
#pragma unroll
  for (int r = 0; r < 8; ++r) {
    const int m = m0 + r + (hiHalf ? 8 : 0);
    z[(size_t)m * D2 + n] = c[r] + bv;
  }
}

// ---------------------------------------------------------------------------
// Phase 4: row L2 norms of z. 1 wave per row; shuffle tree (deterministic).
// ---------------------------------------------------------------------------
__global__ __launch_bounds__(256) void norm_kernel(
    const float* __restrict__ z, float* __restrict__ nrm) {
  const int w = threadIdx.x >> 5, lane = threadIdx.x & 31;
  const int rowi = blockIdx.x * 8 + w;
  const float* __restrict__ zr = z + (size_t)rowi * D2;
  float s = 0.f;
  for (int i = lane; i < D2; i += 32) { float v = zr[i]; s += v * v; }
#pragma unroll
  for (int off = 16; off; off >>= 1) s += __shfl_xor(s, off, 32);
  if (lane == 0) nrm[rowi] = sqrtf(s);
}

// ---------------------------------------------------------------------------
// Phase 5: fused cos-sim (WMMA z @ z^T) + mask + temperature + per-row
// log-sum-exp + positive-pair pick. One block per 16-row band (32 blocks).
// Each wave does 4 column tiles; fixed-order shuffle + LDS reductions.
// Note: cos/T <= 10, so direct log(sum(exp)) is safe; diagonal NEG_INF/T
// underflows exp to exactly 0 (matches masking).
// ---------------------------------------------------------------------------
__global__ __launch_bounds__(256) void sim_kernel(
    const float* __restrict__ z, const float* __restrict__ nrm,
    float* __restrict__ row_nll) {
  __shared__ float esum_sh[8][16];
  __shared__ float pos_sh[8][16];
  const int w = threadIdx.x >> 5, lane = threadIdx.x & 31;
  const int m0 = blockIdx.x * 16;
  const bool hiHalf = lane >= 16;
  const int row = lane & 15;
  const int koff = hiHalf ? 2 : 0;

  float normM[8];
#pragma unroll
  for (int r = 0; r < 8; ++r) normM[r] = nrm[m0 + r + (hiHalf ? 8 : 0)];

  const float* __restrict__ Arow = z + (size_t)(m0 + row) * D2 + koff;
  float esum[8], pos[8];
#pragma unroll
  for (int r = 0; r < 8; ++r) { esum[r] = 0.f; pos[r] = 0.f; }

  for (int ct = w; ct < 32; ct += 8) {
    const int n0 = ct * 16;
    const int ncol = n0 + row;                        // this lane's column
    const float* __restrict__ Brow = z + (size_t)ncol * D2 + koff;
    const float normN = nrm[ncol];
    v8f c = {};
#pragma unroll 2
    for (int k = 0; k < D2; k += 4) {
      v2f a = *(const v2f*)(Arow + k);
      v2f b = *(const v2f*)(Brow + k);
      c = wmma_k4(a, b, c);
    }
#pragma unroll
    for (int r = 0; r < 8; ++r) {
      const int m = m0 + r + (hiHalf ? 8 : 0);
      float denom = fmaxf(normM[r] * normN, EPSV);
      float val = c[r] / denom;
      if (m == ncol) val = NEG_INF_V;
      val *= INV_T;
      esum[r] += expf(val);
      pos[r] += (ncol == (m ^ (Bsz / 2))) ? val : 0.f;
    }
  }

  // reduce across each 16-lane half (xor offsets < 16 keep halves separate)
#pragma unroll
  for (int off = 8; off; off >>= 1) {
#pragma unroll
    for (int r = 0; r < 8; ++r) {
      esum[r] += __shfl_xor(esum[r], off, 32);
      pos[r]  += __shfl_xor(pos[r], off, 32);
    }
  }
  if (lane == 0) {
#pragma unroll
    for (int r = 0; r < 8; ++r) { esum_sh[w][r] = esum[r]; pos_sh[w][r] = pos[r]; }
  }
  if (lane == 16) {
#pragma unroll
    for (int r = 0; r < 8; ++r) { esum_sh[w][8 + r] = esum[r]; pos_sh[w][8 + r] = pos[r]; }
  }
  __syncthreads();
  if (threadIdx.x < 16) {
    float es = 0.f, ps = 0.f;
#pragma unroll
    for (int ww = 0; ww < 8; ++ww) { es += esum_sh[ww][threadIdx.x]; ps += pos_sh[ww][threadIdx.x]; }
    row_nll[m0 + threadIdx.x] = -ps + logf(es);
  }
}

// ---------------------------------------------------------------------------
// Phase 6: mean over 512 row NLLs -> scalar.
// ---------------------------------------------------------------------------
__global__ __launch_bounds__(256) void final_kernel(
    const float* __restrict__ row_nll, float* __restrict__ out) {
  __shared__ float sh[256];
  const int t = threadIdx.x;
  sh[t] = row_nll[t] + row_nll[t + 256];
  __syncthreads();
  for (int off = 128; off; off >>= 1) {
    if (t < off) sh[t] += sh[t + off];
    __syncthreads();
  }
  if (t == 0) out[0] = sh[0] / (float)Bsz;
}

// ---------------------------------------------------------------------------
extern "C" void kernel_launch(void* const* d_in, const int* in_sizes, int n_in,
                              void* d_out, int out_size, void* d_ws, size_t ws_size,
                              hipStream_t stream) {
  const float* x   = (const float*)d_in[0];
  const float* W1c = (const float*)d_in[1];
  const float* b1c = (const float*)d_in[2];
  const float* W2c = (const float*)d_in[3];
  const float* b2c = (const float*)d_in[4];
  const float* W1a = (const float*)d_in[5];
  const float* b1a = (const float*)d_in[6];
  const float* W2a = (const float*)d_in[7];
  const float* b2a = (const float*)d_in[8];
  float* out = (float*)d_out;

  float* ws      = (float*)d_ws;
  float* partial = ws;                                   // 4*512*960
  float* pooled  = partial + (size_t)SSPLIT * Bsz * Dsz; // 512*960
  float* h       = pooled + (size_t)Bsz * Dsz;           // 512*480
  float* z       = h + (size_t)Bsz * D1;                 // 512*240
  float* nrm     = z + (size_t)Bsz * D2;                 // 512
  float* row_nll = nrm + Bsz;                            // 512

  pool_partial_kernel<<<dim3(Bsz, SSPLIT), 240, 0, stream>>>(x, partial);
  pool_reduce_kernel<<<(Bsz * Dsz / 4) / 256, 256, 0, stream>>>(partial, pooled);
  mlp1_kernel<<<120, 256, 0, stream>>>(pooled, W1c, b1c, W1a, b1a, h);
  mlp2_kernel<<<60, 256, 0, stream>>>(h, W2c, b2c, W2a, b2a, z);
  norm_kernel<<<Bsz / 8, 256, 0, stream>>>(z, nrm);
  sim_kernel<<<Bsz / 16, 256, 0, stream>>>(z, nrm, row_nll);
  final_kernel<<<1, 256, 0, stream>>>(row_nll, out);
}